// QuantumLayer_65481071409058
// MI455X (gfx1250) — compile-verified
//
#include <hip/hip_runtime.h>
#include <hip/hip_bf16.h>

typedef __attribute__((ext_vector_type(2))) float v2f;
typedef __attribute__((ext_vector_type(8))) float v8f;

#define NQ      18
#define NLAYERS 2
#define NBATCH  16
#define NSTATE  (1u << 18)   // 2^18 amplitudes per batch element
#define NSLAB   (1u << 14)   // low-group slab: bits 13:0

// Padded LDS addressing: 4 floats of pad per 64-float row.
// phys(i) = i + 4*(i>>6); row base 68*t is 16B-aligned (272B = 17*16B).
#define LDSPAD(i) ((i) + 4u * ((i) >> 6))
#define LDS_FLOATS (NSLAB + 4u * (NSLAB >> 6))   // 17408 floats = 68 KB

// ---------------------------------------------------------------------------
// K0: layer 0 in closed form. |0..0> -> (x RYq) -> product state
//   amp(v) = prod_q (bit_{17-q}(v) ? sin : cos)(theta_q/2),
// then the full CNOT ladder is the basis permutation
//   sigma(v): bit_j' = XOR(bits j..17)  (18-bit suffix XOR).
// Writes state[b][sigma(v)] = amp(v). Also zeroes d_out.
// Blocks of 256 never straddle a batch boundary (2^18 per batch).
// ---------------------------------------------------------------------------
__global__ void qinit1_kernel(const float* __restrict__ angles,
                              float* __restrict__ state,
                              float* __restrict__ out) {
  __shared__ float cc[NQ], sc[NQ];
  const unsigned i = blockIdx.x * 256 + threadIdx.x;
  const unsigned b = i >> 18;
  const unsigned v = i & (NSTATE - 1u);
  if (threadIdx.x < NQ) {
    const float ha = 0.5f * angles[(b * NLAYERS + 0) * NQ + threadIdx.x];
    cc[threadIdx.x] = cosf(ha);
    sc[threadIdx.x] = sinf(ha);
  }
  __syncthreads();

  float r = 1.0f;
#pragma unroll
  for (int q = 0; q < NQ; ++q)
    r *= ((v >> (17 - q)) & 1u) ? sc[q] : cc[q];

  unsigned y = v;                       // 18-bit suffix XOR
  y ^= y >> 1; y ^= y >> 2; y ^= y >> 4; y ^= y >> 8; y ^= y >> 16;
  y &= NSTATE - 1u;

  state[((size_t)b << 18) | y] = r;     // permutes only low 5 bits per 32-lane
  if (i < NBATCH * NQ) out[i] = 0.0f;
}

// ---------------------------------------------------------------------------
// K1: "cross" pass for layer 1. Gates on qubits 0..3 (state bits 17:14):
//   M = C(q2,q3)*C(q1,q2)*C(q0,q1)*(RY0 x RY1 x RY2 x RY3)  (16x16)
// new[g,x] = sum_h M[g,h]*old[h,x] : a 16 x 16 x 2^14 fp32 GEMM per batch
// -> V_WMMA_F32_16X16X4_F32, K=16 in 4 chained steps. One wave per 16x16 tile
// (in-place safe: tile columns owned exclusively by that wave).
// grid = (16 batches, 128), block = 256 (8 waves).
// ---------------------------------------------------------------------------
__global__ void qcross_kernel(const float* __restrict__ angles,
                              float* __restrict__ state, int layer) {
  __shared__ float M[16][16];
  const int b   = blockIdx.x;
  const int tid = threadIdx.x;

  {
    const int k = tid >> 4, h = tid & 15;
    float r = 1.0f;
#pragma unroll
    for (int q = 0; q < 4; ++q) {
      const float ha = 0.5f * angles[(b * NLAYERS + layer) * NQ + q];
      const float c = cosf(ha), s = sinf(ha);
      const int kq = (k >> (3 - q)) & 1;
      const int hq = (h >> (3 - q)) & 1;
      r *= (kq == hq) ? c : (kq ? s : -s);   // RY = [[c,-s],[s,c]]
    }
    int g = k;                 // ladder permutation on bits 3..0 of g
    if (g & 8) g ^= 4;
    if (g & 4) g ^= 2;
    if (g & 2) g ^= 1;
    M[g][h] = r;
  }
  __syncthreads();

  const int wave = tid >> 5;
  const int lane = tid & 31;
  const int tile = blockIdx.y * 8 + wave;        // 0..1023
  const int x0   = tile * 16;
  const int lo   = lane & 15;                    // = m for A, = n for B/D
  const int khi  = lane >> 4;

  float* S = state + (size_t)b * NSTATE;

  v8f d = {};
#pragma unroll
  for (int j = 0; j < 4; ++j) {                  // K = 16 in 4 steps of 4
    const int r0 = 4 * j + 2 * khi;
    v2f a, bb;
    a.x  = M[lo][r0];                            // A[m][k]: lane=m+16*(k>>1)
    a.y  = M[lo][r0 + 1];
    bb.x = S[(size_t)r0       * NSLAB + x0 + lo];// B[k][n]: lane=n+16*(k>>1)
    bb.y = S[(size_t)(r0 + 1) * NSLAB + x0 + lo];
    d = __builtin_amdgcn_wmma_f32_16x16x4_f32(false, a, false, bb,
                                              (short)0, d, false, false);
  }
#pragma unroll
  for (int v = 0; v < 8; ++v) {                  // D[m][n]: m=v+8*khi, n=lo
    const int m = v + 8 * khi;
    S[(size_t)m * NSLAB + x0 + lo] = d[v];
  }
}

// 6 RY butterflies on a 64-element register array; bit lb of j, qubit qA-lb.
__device__ __forceinline__ void ry6(float (&r)[64],
                                    const float* __restrict__ cc,
                                    const float* __restrict__ sc, int qA) {
#pragma unroll
  for (int lb = 0; lb < 6; ++lb) {
    const float c = cc[qA - lb], s = sc[qA - lb];
#pragma unroll
    for (int p = 0; p < 32; ++p) {
      const int j0 = ((p >> lb) << (lb + 1)) | (p & ((1 << lb) - 1));
      const int j1 = j0 | (1 << lb);
      const float a0 = r[j0], a1 = r[j1];
      r[j0] = c * a0 - s * a1;
      r[j1] = s * a0 + c * a1;
    }
  }
}

// ---------------------------------------------------------------------------
// K2: layer-1 "local" pass + fused output reduction.
// One WG per (batch b, high-group h); slab of 2^14 floats in padded LDS.
//  Stage A: bits[5:0] per-thread in VGPRs  -> RY q17..q12 (no LDS traffic)
//  Stage B: bits[13:8] per-thread in VGPRs -> RY q4..q9
//  2 LDS pair sweeps for bits 6,7          -> RY q11,q10
// Then cond-CNOT(q3,q4) + ladder q4..q16 = sigma(v)=suffixXOR14(v)^(h0*0x3FFF)
// is fused into the Z-expectation reduction (never materialized):
//  final index = (h<<14)|sigma(v); bits 17:14 give constant signs (+-T).
// ---------------------------------------------------------------------------
__global__ void qlocal_reduce_kernel(const float* __restrict__ angles,
                                     const float* __restrict__ state,
                                     float* __restrict__ out) {
  extern __shared__ float s0[];                  // LDS_FLOATS floats (68 KB)
  __shared__ float cc[NQ], sc[NQ];
  __shared__ float red[8][15];
  const int b = blockIdx.x, h = blockIdx.y, tid = threadIdx.x;
  const float* G = state + (size_t)b * NSTATE + (size_t)h * NSLAB;

  if (tid < NQ) {
    const float ha = 0.5f * angles[(b * NLAYERS + 1) * NQ + tid];
    cc[tid] = cosf(ha);
    sc[tid] = sinf(ha);
  }
  // coalesced load, padded LDS store (phys stays 16B-aligned per float4)
  for (unsigned i4 = tid; i4 < NSLAB / 4; i4 += 256) {
    const unsigned i = i4 * 4;
    ((float4*)s0)[LDSPAD(i) >> 2] = ((const float4*)G)[i4];
  }
  __syncthreads();

  float r[64];
  {  // Stage A: i = tid*64 + j, phys = 68*tid + j (b128-friendly)
    float4* base = (float4*)(s0 + 68u * tid);
#pragma unroll
    for (int k = 0; k < 16; ++k) {
      float4 t4 = base[k];
      r[4*k] = t4.x; r[4*k+1] = t4.y; r[4*k+2] = t4.z; r[4*k+3] = t4.w;
    }
    ry6(r, cc, sc, 17);                          // qubits 17..12 (bits 0..5)
#pragma unroll
    for (int k = 0; k < 16; ++k) {
      float4 t4 = { r[4*k], r[4*k+1], r[4*k+2], r[4*k+3] };
      base[k] = t4;
    }
  }
  __syncthreads();
  {  // Stage B: i = j*256 + tid, phys = 272*j + tid + 4*(tid>>6) (conflict-free)
    const unsigned offB = tid + 4u * (tid >> 6);
#pragma unroll
    for (int j = 0; j < 64; ++j) r[j] = s0[272u * j + offB];
    ry6(r, cc, sc, 9);                           // qubits 9..4 (bits 8..13)
#pragma unroll
    for (int j = 0; j < 64; ++j) s0[272u * j + offB] = r[j];
  }
  __syncthreads();
  for (int t = 6; t <= 7; ++t) {                 // qubits 11,10 (bits 6,7)
    const float c = cc[17 - t], s = sc[17 - t];
    const unsigned mask = (1u << t) - 1u;
    for (unsigned p = tid; p < NSLAB / 2; p += 256) {
      const unsigned i0 = ((p >> t) << (t + 1)) | (p & mask);
      const unsigned i1 = i0 | (1u << t);
      const unsigned p0 = LDSPAD(i0), p1 = LDSPAD(i1);
      const float a0 = s0[p0], a1 = s0[p1];
      s0[p0] = c * a0 - s * a1;
      s0[p1] = s * a0 + c * a1;
    }
    __syncthreads();
  }

  // fused permutation + Z reduction: acc[j] = sum (+-p) by bit j of sigma(v);
  // acc[14] = total probability (signs for qubits 0..3 come from h).
  const unsigned hmask = (h & 1) ? 0x3FFFu : 0u;
  float acc[15];
#pragma unroll
  for (int j = 0; j < 15; ++j) acc[j] = 0.0f;
  for (unsigned v = tid; v < NSLAB; v += 256) {
    const float a = s0[LDSPAD(v)];
    const float p = a * a;
    unsigned y = v;                              // 14-bit suffix XOR
    y ^= y >> 1; y ^= y >> 2; y ^= y >> 4; y ^= y >> 8;
    y = (y ^ hmask) & 0x3FFFu;
    acc[14] += p;
#pragma unroll
    for (int j = 0; j < 14; ++j)
      acc[j] += ((y >> j) & 1u) ? -p : p;
  }
#pragma unroll
  for (int j = 0; j < 15; ++j)
    for (int off = 16; off > 0; off >>= 1)
      acc[j] += __shfl_down(acc[j], off, 32);
  const int wave = tid >> 5, lane = tid & 31;
  if (lane == 0)
#pragma unroll
    for (int j = 0; j < 15; ++j) red[wave][j] = acc[j];
  __syncthreads();

  if (tid < NQ) {
    float val;
    if (tid < 4) {                               // qubit tid <-> bit (3-tid) of h
      float T = 0.0f;
      for (int w = 0; w < 8; ++w) T += red[w][14];
      val = ((h >> (3 - tid)) & 1) ? -T : T;
    } else {                                     // qubit tid <-> sigma-bit 17-tid
      const int j = 17 - tid;
      float t = 0.0f;
      for (int w = 0; w < 8; ++w) t += red[w][j];
      val = t;
    }
    atomicAdd(&out[b * NQ + tid], val);
  }
}

// ---------------------------------------------------------------------------
extern "C" void kernel_launch(void* const* d_in, const int* in_sizes, int n_in,
                              void* d_out, int out_size, void* d_ws, size_t ws_size,
                              hipStream_t stream) {
  (void)in_sizes; (void)n_in; (void)out_size; (void)ws_size;
  const float* angles = (const float*)d_in[0];   // [16, 2, 18] fp32
  float* out   = (float*)d_out;                  // [16, 18] fp32
  float* state = (float*)d_ws;                   // 16 * 2^18 fp32 = 16 MB scratch

  // Layer 0 (closed form product state + ladder permutation), zero d_out.
  qinit1_kernel<<<(NBATCH * NSTATE) / 256, 256, 0, stream>>>(angles, state, out);
  // Layer 1 cross gates (qubits 0..3): WMMA GEMM pass.
  qcross_kernel<<<dim3(NBATCH, 128), 256, 0, stream>>>(angles, state, 1);
  // Layer 1 local gates + ladder permutation fused with <Z> reduction.
  qlocal_reduce_kernel<<<dim3(NBATCH, 16), 256, LDS_FLOATS * sizeof(float),
                         stream>>>(angles, state, out);
}